// DCRNNCell_74431783240178
// MI455X (gfx1250) — compile-verified
//
#include <hip/hip_runtime.h>
#include <hip/hip_bf16.h>

typedef __attribute__((ext_vector_type(2))) float v2f;
typedef __attribute__((ext_vector_type(8))) float v8f;

#define Bn 4
#define Nn 10000
#define Dd 64
#define Hh 128
#define Cc 192          // D + H
#define Ee 160000
#define EN (Ee + Nn)    // edges + self loops

// padded LDS strides (floats) to avoid 64-bank alignment conflicts
#define SX  68
#define SA  196
#define SC  132
#define SG  388

// ---------------------------------------------------------------------------
// zero the two scatter accumulation buffers (float4 stores)
// ---------------------------------------------------------------------------
__global__ __launch_bounds__(256) void zero_ws(float4* ws, long n4) {
    long i = (long)blockIdx.x * blockDim.x + threadIdx.x;
    if (i < n4) ws[i] = make_float4(0.f, 0.f, 0.f, 0.f);
}

// ---------------------------------------------------------------------------
// pass 1:   tmp1[b, dst, :] += concat(x,h)[b, src, :]
// one thread = one (edge, batch, 4-channel chunk); float4 gather + 4 atomics
// ---------------------------------------------------------------------------
__global__ __launch_bounds__(256) void scatter_pass1(
    const float* __restrict__ x, const float* __restrict__ h,
    const long long* __restrict__ ei, float* __restrict__ tmp1)
{
    long gid = (long)blockIdx.x * blockDim.x + threadIdx.x;
    int  c4  = (int)(gid % 48);
    long rem = gid / 48;
    int  b   = (int)(rem % Bn);
    long e   = rem / Bn;
    if (e >= (long)EN) return;

    int s, d;
    if (e < Ee) { s = (int)ei[e]; d = (int)ei[Ee + e]; }
    else        { s = (int)(e - Ee); d = s; }          // self loop

    int c = c4 * 4;
    float4 v;
    if (c < Dd) v = *(const float4*)(x + ((size_t)b * Nn + s) * Dd + c);
    else        v = *(const float4*)(h + ((size_t)b * Nn + s) * Hh + (c - Dd));

    float* p = tmp1 + ((size_t)b * Nn + d) * Cc + c;
    atomicAdd(p + 0, v.x);
    atomicAdd(p + 1, v.y);
    atomicAdd(p + 2, v.z);
    atomicAdd(p + 3, v.w);
}

// ---------------------------------------------------------------------------
// pass 2:   agg[b, dst, :] += tmp1[b, src, :]
// ---------------------------------------------------------------------------
__global__ __launch_bounds__(256) void scatter_pass2(
    const float* __restrict__ tmp1, const long long* __restrict__ ei,
    float* __restrict__ agg)
{
    long gid = (long)blockIdx.x * blockDim.x + threadIdx.x;
    int  c4  = (int)(gid % 48);
    long rem = gid / 48;
    int  b   = (int)(rem % Bn);
    long e   = rem / Bn;
    if (e >= (long)EN) return;

    int s, d;
    if (e < Ee) { s = (int)ei[e]; d = (int)ei[Ee + e]; }
    else        { s = (int)(e - Ee); d = s; }

    int c = c4 * 4;
    float4 v = *(const float4*)(tmp1 + ((size_t)b * Nn + s) * Cc + c);

    float* p = agg + ((size_t)b * Nn + d) * Cc + c;
    atomicAdd(p + 0, v.x);
    atomicAdd(p + 1, v.y);
    atomicAdd(p + 2, v.z);
    atomicAdd(p + 3, v.w);
}

// ---------------------------------------------------------------------------
// fused:  conv = agg @ lin_w^T + lin_b
//         gi   = x @ w_ih^T + b_ih ,  gh = conv @ w_hh^T + b_hh
//         GRU gate math, store final (16 x 128) tile.
// 256 threads = 8 waves per 16-row tile; all GEMMs via v_wmma_f32_16x16x4_f32.
// ---------------------------------------------------------------------------
__global__ __launch_bounds__(256) void fused_gemm_gru(
    const float* __restrict__ x,     const float* __restrict__ agg,
    const float* __restrict__ lin_w, const float* __restrict__ lin_b,
    const float* __restrict__ w_ih,  const float* __restrict__ w_hh,
    const float* __restrict__ b_ih,  const float* __restrict__ b_hh,
    float* __restrict__ out)
{
    __shared__ float sX[16 * SX];
    __shared__ float sA[16 * SA];
    __shared__ float sConv[16 * SC];
    __shared__ float sGi[16 * SG];
    __shared__ float sGh[16 * SG];

    int tile   = blockIdx.x;              // 0 .. B*625-1
    int b      = tile / 625;
    int n0row  = (tile - b * 625) * 16;
    size_t rowbase = (size_t)b * Nn + n0row;

    int tid  = threadIdx.x;
    int lane = tid & 31;
    int wave = tid >> 5;
    int r    = lane & 15;                 // column / row-in-tile index
    int half = lane >> 4;                 // 0: lanes 0-15, 1: lanes 16-31

    // ---- stage A-tiles into LDS -------------------------------------------
    #pragma unroll
    for (int i = 0; i < 4; ++i) {         // 16*64 floats
        int e = tid + 256 * i;
        int row = e >> 6, col = e & 63;
        sX[row * SX + col] = x[(rowbase + row) * Dd + col];
    }
    #pragma unroll
    for (int i = 0; i < 12; ++i) {        // 16*192 floats
        int e = tid + 256 * i;
        int row = e / Cc, col = e - row * Cc;
        sA[row * SA + col] = agg[(rowbase + row) * Cc + col];
    }
    __syncthreads();

    // ---- conv = sA(16x192) @ lin_w^T + lin_b : one 16-col tile per wave ----
    {
        int n0 = wave * 16;
        float bias = lin_b[n0 + r];
        v8f acc = {bias, bias, bias, bias, bias, bias, bias, bias};
        for (int k0 = 0; k0 < Cc; k0 += 4) {
            int kk = k0 + 2 * half;
            v2f a;  a.x = sA[r * SA + kk];  a.y = sA[r * SA + kk + 1];
            const float* wp = lin_w + (size_t)(n0 + r) * Cc + kk;
            v2f bf; bf.x = wp[0]; bf.y = wp[1];
            acc = __builtin_amdgcn_wmma_f32_16x16x4_f32(
                      false, a, false, bf, (short)0, acc, false, false);
        }
        #pragma unroll
        for (int j = 0; j < 8; ++j)
            sConv[(j + 8 * half) * SC + n0 + r] = acc[j];
    }
    __syncthreads();

    // ---- gi / gh : 24 column tiles, 3 per wave ----------------------------
    for (int t = 0; t < 3; ++t) {
        int n0 = (wave + 8 * t) * 16;
        {   // gi = sX(16x64) @ w_ih^T + b_ih
            float bias = b_ih[n0 + r];
            v8f acc = {bias, bias, bias, bias, bias, bias, bias, bias};
            for (int k0 = 0; k0 < Dd; k0 += 4) {
                int kk = k0 + 2 * half;
                v2f a;  a.x = sX[r * SX + kk];  a.y = sX[r * SX + kk + 1];
                const float* wp = w_ih + (size_t)(n0 + r) * Dd + kk;
                v2f bf; bf.x = wp[0]; bf.y = wp[1];
                acc = __builtin_amdgcn_wmma_f32_16x16x4_f32(
                          false, a, false, bf, (short)0, acc, false, false);
            }
            #pragma unroll
            for (int j = 0; j < 8; ++j)
                sGi[(j + 8 * half) * SG + n0 + r] = acc[j];
        }
        {   // gh = sConv(16x128) @ w_hh^T + b_hh
            float bias = b_hh[n0 + r];
            v8f acc = {bias, bias, bias, bias, bias, bias, bias, bias};
            for (int k0 = 0; k0 < Hh; k0 += 4) {
                int kk = k0 + 2 * half;
                v2f a;  a.x = sConv[r * SC + kk];  a.y = sConv[r * SC + kk + 1];
                const float* wp = w_hh + (size_t)(n0 + r) * Hh + kk;
                v2f bf; bf.x = wp[0]; bf.y = wp[1];
                acc = __builtin_amdgcn_wmma_f32_16x16x4_f32(
                          false, a, false, bf, (short)0, acc, false, false);
            }
            #pragma unroll
            for (int j = 0; j < 8; ++j)
                sGh[(j + 8 * half) * SG + n0 + r] = acc[j];
        }
    }
    __syncthreads();

    // ---- GRU gate math + store (16 x 128 outputs, 8 per thread) -----------
    #pragma unroll
    for (int i = 0; i < 8; ++i) {
        int idx = tid + 256 * i;          // 0 .. 2047
        int m = idx >> 7, n = idx & 127;
        float gir = sGi[m * SG + n],        ghr = sGh[m * SG + n];
        float giz = sGi[m * SG + 128 + n],  ghz = sGh[m * SG + 128 + n];
        float gin = sGi[m * SG + 256 + n],  ghn = sGh[m * SG + 256 + n];
        float rr = 1.f / (1.f + __expf(-(gir + ghr)));
        float zz = 1.f / (1.f + __expf(-(giz + ghz)));
        float nn = tanhf(gin + rr * ghn);
        float co = sConv[m * SC + n];
        out[(rowbase + m) * Hh + n] = (1.f - zz) * nn + zz * co;
    }
}

// ---------------------------------------------------------------------------
extern "C" void kernel_launch(void* const* d_in, const int* in_sizes, int n_in,
                              void* d_out, int out_size, void* d_ws, size_t ws_size,
                              hipStream_t stream)
{
    const float*     x     = (const float*)d_in[0];
    const float*     h     = (const float*)d_in[1];
    const long long* ei    = (const long long*)d_in[2];
    const float*     lin_w = (const float*)d_in[3];
    const float*     lin_b = (const float*)d_in[4];
    const float*     w_ih  = (const float*)d_in[5];
    const float*     w_hh  = (const float*)d_in[6];
    const float*     b_ih  = (const float*)d_in[7];
    const float*     b_hh  = (const float*)d_in[8];
    float*           out   = (float*)d_out;

    float* tmp1 = (float*)d_ws;                       // (B,N,192)
    float* agg  = tmp1 + (size_t)Bn * Nn * Cc;        // (B,N,192)

    // zero both accumulation buffers (fresh every call: deterministic)
    long n4 = (long)2 * Bn * Nn * Cc / 4;
    zero_ws<<<(unsigned)((n4 + 255) / 256), 256, 0, stream>>>((float4*)d_ws, n4);

    long total  = (long)EN * Bn * 48;                 // edge * batch * chan4
    unsigned blocks = (unsigned)((total + 255) / 256);
    scatter_pass1<<<blocks, 256, 0, stream>>>(x, h, ei, tmp1);
    scatter_pass2<<<blocks, 256, 0, stream>>>(tmp1, ei, agg);

    fused_gemm_gru<<<Bn * 625, 256, 0, stream>>>(
        x, agg, lin_w, lin_b, w_ih, w_hh, b_ih, b_hh, out);
}